// ConstantBaseLine_74663711474121
// MI455X (gfx1250) — compile-verified
//
#include <hip/hip_runtime.h>
#include <stdint.h>

// Problem constants from the reference (fixed shapes).
#define B_ROWS  64
#define T_LEN   500000
#define TPB     256
#define ITEMS   16
#define CHUNK   (TPB * ITEMS)                       // 4096 elements per block
#define NCHUNKS ((T_LEN + CHUNK - 1) / CHUNK)       // 123 chunks per row
#define NWAVES  (TPB / 32)

// ---------------------------------------------------------------------------
// gfx1250 async copy: GLOBAL memory -> LDS, 16 bytes per lane, ASYNCcnt path.
// GVS addressing: memaddr = SGPR64 + VGPR32 + inst_offset.
// VDST vgpr holds the LDS byte address (low 32 bits of the generic pointer).
// ---------------------------------------------------------------------------
__device__ __forceinline__ void async_load_b128_to_lds(unsigned lds_byte,
                                                       unsigned gbyte,
                                                       const void* sbase) {
  asm volatile("global_load_async_to_lds_b128 %0, %1, %2"
               :
               : "v"(lds_byte), "v"(gbyte), "s"(sbase)
               : "memory");
}

__device__ __forceinline__ void wait_async0() {
  asm volatile("s_wait_asynccnt 0" ::: "memory");
}

// ---------------------------------------------------------------------------
// Pass 1: per-chunk aggregate = last dry (row-relative) index in chunk, or -1.
// Reads only the wet flags (1 byte/elem).
// ---------------------------------------------------------------------------
__global__ __launch_bounds__(TPB) void k_chunk_agg(
    const unsigned char* __restrict__ wet, int* __restrict__ ws_idx) {
  const int bx  = blockIdx.x;
  const int row = bx / NCHUNKS;
  const int ch  = bx % NCHUNKS;
  const size_t rowbase = (size_t)row * T_LEN;
  const int tid = threadIdx.x;
  const int seg = ch * CHUNK + tid * ITEMS;   // row-relative start of this thread's run

  union { uint4 v; unsigned char b[ITEMS]; } u;
  if (seg + ITEMS <= T_LEN) {                 // 16B-aligned (seg % 16 == 0, T % 16 == 0)
    u.v = *reinterpret_cast<const uint4*>(wet + rowbase + seg);
  } else {
#pragma unroll
    for (int j = 0; j < ITEMS; ++j)
      u.b[j] = (seg + j < T_LEN) ? wet[rowbase + seg + j] : (unsigned char)1;
  }

  int lidx = -1;
#pragma unroll
  for (int j = 0; j < ITEMS; ++j) {
    const int g = seg + j;
    if (g < T_LEN && (g == 0 || u.b[j] == 0)) lidx = g;   // index 0 forced dry
  }

  // wave32 max reduce, then cross-wave via LDS
#pragma unroll
  for (int d = 16; d >= 1; d >>= 1) lidx = max(lidx, __shfl_xor(lidx, d, 32));

  __shared__ int wtot[NWAVES];
  const int lane = tid & 31, wid = tid >> 5;
  if (lane == 0) wtot[wid] = lidx;
  __syncthreads();
  if (tid == 0) {
    int m = wtot[0];
#pragma unroll
    for (int w = 1; w < NWAVES; ++w) m = max(m, wtot[w]);
    ws_idx[bx] = m;
  }
}

// ---------------------------------------------------------------------------
// Pass 2: per-row exclusive running-max over chunk aggregates. carry[r][0] = 0
// (index 0 forced dry). 64 rows x 123 chunks -> trivial.
// ---------------------------------------------------------------------------
__global__ void k_carry_scan(const int* __restrict__ ws_idx,
                             int* __restrict__ carry) {
  const int r = threadIdx.x;
  if (r >= B_ROWS) return;
  int run = 0;
  for (int c = 0; c < NCHUNKS; ++c) {
    carry[r * NCHUNKS + c] = run;
    run = max(run, ws_idx[r * NCHUNKS + c]);
  }
}

// ---------------------------------------------------------------------------
// Pass 3: apply. Async-copy the 16KB att tile into LDS while computing the
// per-thread last-dry index from wet flags; wave + block exclusive max-scan
// gives each thread the most recent dry index before its run; one gather seeds
// the hold value; 16 contiguous outputs per thread via float4 stores.
// ---------------------------------------------------------------------------
__global__ __launch_bounds__(TPB) void k_apply(
    const float* __restrict__ att, const unsigned char* __restrict__ wet,
    const int* __restrict__ carry, float* __restrict__ out) {
  const int bx  = blockIdx.x;
  const int row = bx / NCHUNKS;
  const int ch  = bx % NCHUNKS;
  const size_t rowbase = (size_t)row * T_LEN;
  const int tid    = threadIdx.x;
  const int cstart = ch * CHUNK;
  const int seg    = cstart + tid * ITEMS;

  __shared__ float tile[CHUNK];
  __shared__ int   wtot[NWAVES];

  // Issue async tile copy first (4 x b128 per lane = 64B), overlapped below.
  {
    const float* sbase = att + rowbase;                    // block-uniform -> SGPR pair
    const unsigned lds0 = (unsigned)(uintptr_t)&tile[0];   // LDS byte offset
#pragma unroll
    for (int k = 0; k < 4; ++k) {
      int gElem = cstart + tid * ITEMS + k * 4;
      if (gElem > T_LEN - 4) gElem = T_LEN - 4;            // in-bounds, 16B-aligned clamp
      async_load_b128_to_lds(lds0 + (unsigned)((tid * ITEMS + k * 4) * 4),
                             (unsigned)(gElem * 4), sbase);
    }
  }

  // Wet flags for this thread's run.
  union { uint4 v; unsigned char b[ITEMS]; } u;
  if (seg + ITEMS <= T_LEN) {
    u.v = *reinterpret_cast<const uint4*>(wet + rowbase + seg);
  } else {
#pragma unroll
    for (int j = 0; j < ITEMS; ++j)
      u.b[j] = (seg + j < T_LEN) ? wet[rowbase + seg + j] : (unsigned char)1;
  }

  int lidx = -1;
#pragma unroll
  for (int j = 0; j < ITEMS; ++j) {
    const int g = seg + j;
    if (g < T_LEN && (g == 0 || u.b[j] == 0)) lidx = g;
  }

  // Wave32 inclusive max-scan (5 shfl_up steps).
  const int lane = tid & 31, wid = tid >> 5;
  int incl = lidx;
#pragma unroll
  for (int d = 1; d < 32; d <<= 1) {
    const int n = __shfl_up(incl, d, 32);
    if (lane >= d) incl = max(incl, n);
  }
  if (lane == 31) wtot[wid] = incl;

  wait_async0();      // this wave's async LDS writes complete
  __syncthreads();    // tile + wtot visible block-wide

  // Exclusive prefix: block carry + earlier waves + earlier lanes.
  int prefix = carry[bx];
#pragma unroll
  for (int w = 0; w < NWAVES; ++w)
    if (w < wid) prefix = max(prefix, wtot[w]);
  const int up = __shfl_up(incl, 1, 32);
  if (lane > 0) prefix = max(prefix, up);
  // prefix in [0, T): the most recent dry index before this thread's run.

  float cur = att[rowbase + prefix];   // seed gather (mostly L2-resident)

  const int lbase = tid * ITEMS;
  if (seg + ITEMS <= T_LEN) {
    float o[ITEMS];
#pragma unroll
    for (int j = 0; j < ITEMS; ++j) {
      const bool dry = ((seg + j) == 0) || (u.b[j] == 0);
      if (dry) cur = tile[lbase + j];
      o[j] = cur;
    }
    float4* dst = reinterpret_cast<float4*>(out + rowbase + seg);
#pragma unroll
    for (int k = 0; k < 4; ++k)
      dst[k] = make_float4(o[4 * k], o[4 * k + 1], o[4 * k + 2], o[4 * k + 3]);
  } else {
    for (int j = 0; j < ITEMS && seg + j < T_LEN; ++j) {
      const bool dry = ((seg + j) == 0) || (u.b[j] == 0);
      if (dry) cur = tile[lbase + j];
      out[rowbase + seg + j] = cur;
    }
  }
}

// ---------------------------------------------------------------------------
// Launch: pass1 (aggregates) -> pass2 (per-row carry) -> pass3 (apply).
// Workspace: 2 * B * NCHUNKS ints (~63 KB).
// ---------------------------------------------------------------------------
extern "C" void kernel_launch(void* const* d_in, const int* in_sizes, int n_in,
                              void* d_out, int out_size, void* d_ws, size_t ws_size,
                              hipStream_t stream) {
  (void)in_sizes; (void)n_in; (void)out_size; (void)ws_size;
  const float*         att = (const float*)d_in[0];
  const unsigned char* wet = (const unsigned char*)d_in[1];  // jnp.bool_ = 1 byte/elem
  float* out = (float*)d_out;

  int* ws_idx = (int*)d_ws;                       // B*NCHUNKS ints
  int* carry  = ws_idx + B_ROWS * NCHUNKS;        // B*NCHUNKS ints

  const dim3 grid(B_ROWS * NCHUNKS);
  k_chunk_agg<<<grid, TPB, 0, stream>>>(wet, ws_idx);
  k_carry_scan<<<1, B_ROWS, 0, stream>>>(ws_idx, carry);
  k_apply<<<grid, TPB, 0, stream>>>(att, wet, carry, out);
}